// TransformerModel_72267119723342
// MI455X (gfx1250) — compile-verified
//
#include <hip/hip_runtime.h>
#include <math.h>

// ---------------------------------------------------------------------------
// Transformer encoder + greedy decoder for MI455X (gfx1250, wave32, WMMA).
// fp32 end-to-end using V_WMMA_F32_16X16X4_F32 for all encoder-side GEMMs.
// All workspace activation buffers are padded to 208 rows (13 x 16-row tiles)
// so tile loops have NO bounds guards -> straight-line load/wmma codegen.
// ---------------------------------------------------------------------------

typedef __attribute__((ext_vector_type(2))) float v2f;
typedef __attribute__((ext_vector_type(8))) float v8f;

#define S_LEN  200
#define SP_LEN 208          // padded row count (13 tiles of 16)
#define EMB    64
#define FFDIM  2048
#define VOCAB  200
#define NH     4
#define DH     16
#define STEPS  100
#define MT     13           // SP_LEN / 16

__device__ __forceinline__ v8f wmma_f32(v2f a, v2f b, v8f c) {
  // D = A(16x4) * B(4x16) + C, fp32.
  // (neg_a, A, neg_b, B, c_mod, C, reuse_a, reuse_b)
  return __builtin_amdgcn_wmma_f32_16x16x4_f32(false, a, false, b, (short)0, c,
                                               false, false);
}

// ---------------------------------------------------------------------------
// h = src @ lin_w^T + lin_b (200x3 @ 3x64); pad rows [200,208) with zeros.
// ---------------------------------------------------------------------------
__global__ void k_embed(const float* __restrict__ src,
                        const float* __restrict__ Wl,
                        const float* __restrict__ bl,
                        float* __restrict__ H) {
  int i = blockIdx.x * blockDim.x + threadIdx.x;
  if (i >= SP_LEN * EMB) return;
  int r = i / EMB, c = i % EMB;
  float acc = 0.f;
  if (r < S_LEN) {
    acc = bl[c];
    #pragma unroll
    for (int k = 0; k < 3; ++k) acc += src[r * 3 + k] * Wl[c * 3 + k];
  }
  H[i] = acc;
}

// ---------------------------------------------------------------------------
// WMMA GEMM, fixed K=N=64: Y(208x64) = X(208x64) @ W^T (+bias, +residual).
// W row-major (64 x 64). One wave per 16x16 tile; K-loop fully unrolled so
// the 32 global_load_b64s pipeline ahead of the 16-deep wmma chain.
// ---------------------------------------------------------------------------
__global__ __launch_bounds__(32) void k_gemm64(const float* __restrict__ X,
                                               const float* __restrict__ W,
                                               const float* __restrict__ bias,
                                               const float* __restrict__ res,
                                               float* __restrict__ Y) {
  const int lane = threadIdx.x & 31;
  const int tm   = blockIdx.x >> 2;          // 13 row tiles
  const int tc   = blockIdx.x & 3;           // 4 col tiles
  const int r16  = lane & 15;
  const int hi   = lane >> 4;                // half-wave: K pair {0,1}/{2,3}
  const float* xp = X + (tm * 16 + r16) * EMB + (hi << 1);
  const float* wp = W + (tc * 16 + r16) * EMB + (hi << 1);

  v8f acc = {};
  #pragma unroll
  for (int k0 = 0; k0 < EMB; k0 += 4) {
    v2f a, b;
    a.x = xp[k0]; a.y = xp[k0 + 1];
    b.x = wp[k0]; b.y = wp[k0 + 1];
    acc = wmma_f32(a, b, acc);
  }
  #pragma unroll
  for (int v = 0; v < 8; ++v) {
    const int rr = tm * 16 + v + 8 * hi;
    const int cc = tc * 16 + r16;
    float val = acc[v] + bias[cc];
    if (res) val += res[rr * EMB + cc];
    Y[rr * EMB + cc] = val;
  }
}

// ---------------------------------------------------------------------------
// Encoder self-attention for one (head, 16-query tile):
//   P = softmax(Q_h K_h^T / 4),  O_h = P V_h.  Scores staged in LDS.
// Q/K/V are padded buffers: no load guards; masking applied to scores only.
// ---------------------------------------------------------------------------
__global__ __launch_bounds__(32) void k_attn(const float* __restrict__ Q,
                                             const float* __restrict__ Kb,
                                             const float* __restrict__ Vb,
                                             float* __restrict__ O) {
  __shared__ float sc[16][SP_LEN];
  const int lane = threadIdx.x & 31;
  const int head = blockIdx.x / MT;
  const int qt   = blockIdx.x % MT;
  const int r16  = lane & 15;
  const int hi   = lane >> 4;
  const int h16  = head * DH;
  const float* qp = Q + (qt * 16 + r16) * EMB + h16 + (hi << 1);

  // Phase 1: scores = Q_h @ K_h^T * 0.25, mask pad keys to -1e30
  for (int kt = 0; kt < MT; ++kt) {
    const float* kp = Kb + (kt * 16 + r16) * EMB + h16 + (hi << 1);
    v8f acc = {};
    #pragma unroll
    for (int k0 = 0; k0 < DH; k0 += 4) {
      v2f a, b;
      a.x = qp[k0]; a.y = qp[k0 + 1];
      b.x = kp[k0]; b.y = kp[k0 + 1];
      acc = wmma_f32(a, b, acc);
    }
    const int kc = kt * 16 + r16;
    #pragma unroll
    for (int v = 0; v < 8; ++v)
      sc[v + 8 * hi][kc] = (kc < S_LEN) ? acc[v] * 0.25f : -1e30f;
  }
  __syncthreads();

  // Phase 2: row softmax; lane pair (l, l^16) splits the 208 columns,
  // combined with wave shuffles. exp(-1e30 - m) == 0 handles the padding.
  {
    const int c0 = hi * (SP_LEN / 2), c1 = c0 + SP_LEN / 2;
    float m = -1e30f;
    for (int j = c0; j < c1; ++j) m = fmaxf(m, sc[r16][j]);
    m = fmaxf(m, __shfl_xor(m, 16, 32));
    float s = 0.f;
    for (int j = c0; j < c1; ++j) { float e = __expf(sc[r16][j] - m); sc[r16][j] = e; s += e; }
    s += __shfl_xor(s, 16, 32);
    const float inv = 1.f / s;
    for (int j = c0; j < c1; ++j) sc[r16][j] *= inv;
  }
  __syncthreads();

  // Phase 3: O_h = P @ V_h  (K = 200, exact multiple of 4; pad cols are 0)
  v8f acc = {};
  #pragma unroll 5
  for (int k0 = 0; k0 < S_LEN; k0 += 4) {
    const int kb = k0 + (hi << 1);
    v2f a, b;
    a.x = sc[r16][kb];
    a.y = sc[r16][kb + 1];
    b.x = Vb[kb * EMB + h16 + r16];
    b.y = Vb[(kb + 1) * EMB + h16 + r16];
    acc = wmma_f32(a, b, acc);
  }
  #pragma unroll
  for (int v = 0; v < 8; ++v)
    O[(qt * 16 + v + 8 * hi) * EMB + h16 + r16] = acc[v];
}

// ---------------------------------------------------------------------------
// Fused FFN per 16-row tile: Y = relu(X W1^T + b1) W2^T + b2 + X (residual).
// Hidden activations staged 16x16 at a time through LDS (never hit HBM).
// ---------------------------------------------------------------------------
__global__ __launch_bounds__(32) void k_ff(const float* __restrict__ X,
                                           const float* __restrict__ W1,
                                           const float* __restrict__ B1,
                                           const float* __restrict__ W2,
                                           const float* __restrict__ B2,
                                           float* __restrict__ Y) {
  __shared__ float hb[16][17];
  const int lane = threadIdx.x & 31;
  const int tm   = blockIdx.x;
  const int r16  = lane & 15;
  const int hi   = lane >> 4;
  const float* xp = X + (tm * 16 + r16) * EMB + (hi << 1);

  v8f acc[4] = {{}, {}, {}, {}};
  for (int c0 = 0; c0 < FFDIM; c0 += 16) {
    // H = relu(X_tile @ W1_chunk^T + b1)
    const float* w1p = W1 + (c0 + r16) * EMB + (hi << 1);
    v8f hacc = {};
    #pragma unroll
    for (int k0 = 0; k0 < EMB; k0 += 4) {
      v2f a, b;
      a.x = xp[k0];  a.y = xp[k0 + 1];
      b.x = w1p[k0]; b.y = w1p[k0 + 1];
      hacc = wmma_f32(a, b, hacc);
    }
    const float bias1 = B1[c0 + r16];
    __syncthreads();                         // WAR vs previous chunk readers
    #pragma unroll
    for (int v = 0; v < 8; ++v)
      hb[v + 8 * hi][r16] = fmaxf(hacc[v] + bias1, 0.f);  // D-layout -> LDS
    __syncthreads();
    // acc[nt] += H(16x16) @ W2^T[c0:c0+16, nt*16 : nt*16+16]
    #pragma unroll
    for (int nt = 0; nt < 4; ++nt) {
      const float* w2p = W2 + (nt * 16 + r16) * FFDIM + c0 + (hi << 1);
      #pragma unroll
      for (int k0 = 0; k0 < 16; k0 += 4) {
        v2f a, b;
        a.x = hb[r16][k0 + (hi << 1)];
        a.y = hb[r16][k0 + (hi << 1) + 1];
        b.x = w2p[k0];
        b.y = w2p[k0 + 1];
        acc[nt] = wmma_f32(a, b, acc[nt]);
      }
    }
  }
  #pragma unroll
  for (int nt = 0; nt < 4; ++nt)
    #pragma unroll
    for (int v = 0; v < 8; ++v) {
      const int rr = tm * 16 + v + 8 * hi;
      const int cc = nt * 16 + r16;
      Y[rr * EMB + cc] = acc[nt][v] + B2[cc] + X[rr * EMB + cc];
    }
}

// ---------------------------------------------------------------------------
// Row LayerNorm over width 64: one wave per row, wave32 shuffle reductions.
// ---------------------------------------------------------------------------
__global__ void k_ln(const float* __restrict__ X, const float* __restrict__ G,
                     const float* __restrict__ B, float* __restrict__ Y,
                     int rows) {
  const int wave = (int)((blockIdx.x * blockDim.x + threadIdx.x) >> 5);
  const int lane = threadIdx.x & 31;
  if (wave >= rows) return;
  const float x0 = X[wave * EMB + lane];
  const float x1 = X[wave * EMB + 32 + lane];
  float s = x0 + x1, ss = x0 * x0 + x1 * x1;
  #pragma unroll
  for (int off = 16; off > 0; off >>= 1) {
    s  += __shfl_xor(s,  off, 32);
    ss += __shfl_xor(ss, off, 32);
  }
  const float mu  = s * (1.f / EMB);
  const float var = ss * (1.f / EMB) - mu * mu;
  const float inv = rsqrtf(var + 1e-5f);
  Y[wave * EMB + lane]      = (x0 - mu) * inv * G[lane]      + B[lane];
  Y[wave * EMB + 32 + lane] = (x1 - mu) * inv * G[lane + 32] + B[lane + 32];
}

// ---------------------------------------------------------------------------
// Persistent greedy decoder: 100 serial steps in one block, state in LDS.
// Self-attn with Sq=Sk=1 collapses to outproj(Vproj(d)).
// ---------------------------------------------------------------------------
__device__ __forceinline__ void ln64(const float* x, const float* g,
                                     const float* b, float* y, float* stats,
                                     int tid) {
  if (tid == 0) {
    float s = 0.f, ss = 0.f;
    for (int k = 0; k < EMB; ++k) { s += x[k]; ss += x[k] * x[k]; }
    const float mu = s * (1.f / EMB);
    stats[0] = mu;
    stats[1] = rsqrtf(ss * (1.f / EMB) - mu * mu + 1e-5f);
  }
  __syncthreads();
  if (tid < EMB) y[tid] = (x[tid] - stats[0]) * stats[1] * g[tid] + b[tid];
  __syncthreads();
}

__global__ __launch_bounds__(256) void k_decoder(
    const float* __restrict__ kmem, const float* __restrict__ vmem,
    const float* __restrict__ sa_in_w, const float* __restrict__ sa_in_b,
    const float* __restrict__ sa_out_w, const float* __restrict__ sa_out_b,
    const float* __restrict__ ln1_g, const float* __restrict__ ln1_b,
    const float* __restrict__ ca_in_w, const float* __restrict__ ca_in_b,
    const float* __restrict__ ca_out_w, const float* __restrict__ ca_out_b,
    const float* __restrict__ ln2_g, const float* __restrict__ ln2_b,
    const float* __restrict__ ff1_w, const float* __restrict__ ff1_b,
    const float* __restrict__ ff2_w, const float* __restrict__ ff2_b,
    const float* __restrict__ ln3_g, const float* __restrict__ ln3_b,
    const float* __restrict__ out_w, const float* __restrict__ out_b,
    float* __restrict__ out) {
  __shared__ float d[EMB], t0[EMB], t1[EMB], qv[EMB];
  __shared__ float sc[NH][S_LEN];
  __shared__ float h1[FFDIM];
  __shared__ float logits[VOCAB];
  __shared__ float mask[VOCAB];
  __shared__ float stats[4];            // [0]=mu [1]=inv [2]=sum(logp)

  const int tid = threadIdx.x;
  if (tid < EMB) d[tid] = 0.f;
  for (int j = tid; j < VOCAB; j += blockDim.x) mask[j] = 0.f;
  if (tid == 0) stats[2] = 0.f;
  __syncthreads();

  for (int t = 0; t < STEPS; ++t) {
    // ---- self-attention (Sq=Sk=1): out = outproj(Vproj(d)) -------------
    if (tid < EMB) {
      float acc = sa_in_b[2 * EMB + tid];
      const float* w = sa_in_w + (2 * EMB + tid) * EMB;
      for (int k = 0; k < EMB; ++k) acc += d[k] * w[k];
      t0[tid] = acc;
    }
    __syncthreads();
    if (tid < EMB) {
      float acc = sa_out_b[tid];
      const float* w = sa_out_w + tid * EMB;
      for (int k = 0; k < EMB; ++k) acc += t0[k] * w[k];
      t1[tid] = d[tid] + acc;
    }
    __syncthreads();
    ln64(t1, ln1_g, ln1_b, d, stats, tid);

    // ---- cross-attention over precomputed memory K/V -------------------
    if (tid < EMB) {
      float acc = ca_in_b[tid];
      const float* w = ca_in_w + tid * EMB;     // Wq block of packed in_proj
      for (int k = 0; k < EMB; ++k) acc += d[k] * w[k];
      qv[tid] = acc;
    }
    __syncthreads();
    for (int idx = tid; idx < NH * S_LEN; idx += blockDim.x) {
      const int h = idx / S_LEN, kk = idx % S_LEN;
      float acc = 0.f;
      const float* kp = kmem + kk * EMB + h * DH;
      const float* qp = qv + h * DH;
      #pragma unroll
      for (int k = 0; k < DH; ++k) acc += qp[k] * kp[k];
      sc[h][kk] = acc * 0.25f;                  // 1/sqrt(16)
    }
    __syncthreads();
    if (tid < NH) {
      float m = -1e30f;
      for (int j = 0; j < S_LEN; ++j) m = fmaxf(m, sc[tid][j]);
      float s = 0.f;
      for (int j = 0; j < S_LEN; ++j) { float e = __expf(sc[tid][j] - m); sc[tid][j] = e; s += e; }
      const float inv = 1.f / s;
      for (int j = 0; j < S_LEN; ++j) sc[tid][j] *= inv;
    }
    __syncthreads();
    if (tid < EMB) {
      const int h = tid / DH, dh = tid % DH;
      float acc = 0.f;
      for (int k = 0; k < S_LEN; ++k) acc += sc[h][k] * vmem[k * EMB + h * DH + dh];
      t0[tid] = acc;
    }
    __syncthreads();
    if (tid < EMB) {
      float acc = ca_out_b[tid];
      const float* w = ca_out_w + tid * EMB;
      for (int k = 0; k < EMB; ++k) acc += t0[k] * w[k];
      t1[tid] = d[tid] + acc;
    }
    __syncthreads();
    ln64(t1, ln2_g, ln2_b, d, stats, tid);

    // ---- feed-forward (dominant per-step work; prefetch weight stream) --
    for (int j = tid; j < FFDIM; j += blockDim.x) {
      const int jn = (j + 256 < FFDIM) ? j + 256 : j;
      __builtin_prefetch(ff1_w + jn * EMB, 0, 0);   // global_prefetch_b8
      float acc = ff1_b[j];
      const float* w = ff1_w + j * EMB;
      for (int k = 0; k < EMB; ++k) acc += d[k] * w[k];
      h1[j] = fmaxf(acc, 0.f);
    }
    __syncthreads();
    if (tid < EMB) {
      float acc = ff2_b[tid];
      const float* w = ff2_w + tid * FFDIM;
      for (int k = 0; k < FFDIM; ++k) acc += h1[k] * w[k];
      t1[tid] = d[tid] + acc;
    }
    __syncthreads();
    ln64(t1, ln3_g, ln3_b, d, stats, tid);

    // ---- output projection, masked softmax, argmax ----------------------
    for (int j = tid; j < VOCAB; j += blockDim.x) {
      float acc = out_b[j] - mask[j] * 1e9f;
      const float* w = out_w + j * EMB;
      for (int k = 0; k < EMB; ++k) acc += d[k] * w[k];
      logits[j] = acc;
    }
    __syncthreads();
    if (tid == 0) {
      int am = 0; float m = logits[0];
      for (int j = 1; j < VOCAB; ++j) if (logits[j] > m) { m = logits[j]; am = j; }
      float s = 0.f;
      for (int j = 0; j < VOCAB; ++j) s += __expf(logits[j] - m);
      stats[2] += logits[am] - m - __logf(s);   // log softmax at argmax
      mask[am] = 1.f;
      out[t] = (float)am;
    }
    __syncthreads();
  }
  if (tid == 0) out[STEPS] = stats[2];
}

// ---------------------------------------------------------------------------
extern "C" void kernel_launch(void* const* d_in, const int* in_sizes, int n_in,
                              void* d_out, int out_size, void* d_ws,
                              size_t ws_size, hipStream_t stream) {
  (void)in_sizes; (void)n_in; (void)out_size; (void)ws_size;
  const float* src          = (const float*)d_in[0];
  const float* lin_w        = (const float*)d_in[1];
  const float* lin_b        = (const float*)d_in[2];
  const float* enc_sa_in_w  = (const float*)d_in[3];
  const float* enc_sa_in_b  = (const float*)d_in[4];
  const float* enc_sa_out_w = (const float*)d_in[5];
  const float* enc_sa_out_b = (const float*)d_in[6];
  const float* enc_ln1_g    = (const float*)d_in[7];
  const float* enc_ln1_b    = (const float*)d_in[8];
  const float* enc_ff1_w    = (const float*)d_in[9];
  const float* enc_ff1_b    = (const float*)d_in[10];
  const float* enc_ff2_w    = (const float*)d_in[11];
  const float* enc_ff2_b    = (const float*)d_in[12];
  const float* enc_ln2_g    = (const float*)d_in[13];
  const float* enc_ln2_b    = (const float*)d_in[14];
  const float* dec_sa_in_w  = (const float*)d_in[15];
  const float* dec_sa_in_b  = (const float*)d_in[16];
  const float* dec_sa_out_w = (const float*)d_in[17];
  const float* dec_sa_out_b = (const float*)d_in[18];
  const float* dec_ln1_g    = (const float*)d_in[19];
  const float* dec_ln1_b    = (const float*)d_in[20];
  const float* dec_ca_in_w  = (const float*)d_in[21];
  const float* dec_ca_in_b  = (const float*)d_in[22];
  const float* dec_ca_out_w = (const float*)d_in[23];
  const float* dec_ca_out_b = (const float*)d_in[24];
  const float* dec_ln2_g    = (const float*)d_in[25];
  const float* dec_ln2_b    = (const float*)d_in[26];
  const float* dec_ff1_w    = (const float*)d_in[27];
  const float* dec_ff1_b    = (const float*)d_in[28];
  const float* dec_ff2_w    = (const float*)d_in[29];
  const float* dec_ff2_b    = (const float*)d_in[30];
  const float* dec_ln3_g    = (const float*)d_in[31];
  const float* dec_ln3_b    = (const float*)d_in[32];
  const float* out_w        = (const float*)d_in[33];
  const float* out_b        = (const float*)d_in[34];

  float* ws = (float*)d_ws;
  const int SE = SP_LEN * EMB;          // 13312 floats per padded buffer
  float* h   = ws + 0 * SE;
  float* q   = ws + 1 * SE;
  float* kbf = ws + 2 * SE;
  float* vbf = ws + 3 * SE;
  float* ao  = ws + 4 * SE;
  float* tb  = ws + 5 * SE;
  float* x   = ws + 6 * SE;
  float* fo  = ws + 7 * SE;
  float* mem = ws + 8 * SE;
  float* km  = ws + 9 * SE;
  float* vm  = ws + 10 * SE;

  const dim3 gTiles(MT * (EMB / 16));   // 52 tile-waves per 208x64 GEMM
  const dim3 gLn((SP_LEN * 32 + 255) / 256);   // 26 blocks x 8 waves

  k_embed<<<(SP_LEN * EMB + 255) / 256, 256, 0, stream>>>(src, lin_w, lin_b, h);

  // Encoder self-attention: QKV projections (packed [Wq;Wk;Wv])
  k_gemm64<<<gTiles, 32, 0, stream>>>(h, enc_sa_in_w,                 enc_sa_in_b,           nullptr, q);
  k_gemm64<<<gTiles, 32, 0, stream>>>(h, enc_sa_in_w + EMB * EMB,     enc_sa_in_b + EMB,     nullptr, kbf);
  k_gemm64<<<gTiles, 32, 0, stream>>>(h, enc_sa_in_w + 2 * EMB * EMB, enc_sa_in_b + 2 * EMB, nullptr, vbf);
  k_attn<<<NH * MT, 32, 0, stream>>>(q, kbf, vbf, ao);
  k_gemm64<<<gTiles, 32, 0, stream>>>(ao, enc_sa_out_w, enc_sa_out_b, h, tb);
  k_ln<<<gLn, 256, 0, stream>>>(tb, enc_ln1_g, enc_ln1_b, x, SP_LEN);

  // Encoder FFN (fused, residual included) + LN2 -> memory
  k_ff<<<MT, 32, 0, stream>>>(x, enc_ff1_w, enc_ff1_b, enc_ff2_w, enc_ff2_b, fo);
  k_ln<<<gLn, 256, 0, stream>>>(fo, enc_ln2_g, enc_ln2_b, mem, SP_LEN);

  // Precompute cross-attention K/V of memory (decoder ca in_proj K,V blocks)
  k_gemm64<<<gTiles, 32, 0, stream>>>(mem, dec_ca_in_w + EMB * EMB,     dec_ca_in_b + EMB,     nullptr, km);
  k_gemm64<<<gTiles, 32, 0, stream>>>(mem, dec_ca_in_w + 2 * EMB * EMB, dec_ca_in_b + 2 * EMB, nullptr, vm);

  // Greedy decode: 100 serial steps in one persistent block
  k_decoder<<<1, 256, 0, stream>>>(km, vm,
      dec_sa_in_w, dec_sa_in_b, dec_sa_out_w, dec_sa_out_b,
      dec_ln1_g, dec_ln1_b,
      dec_ca_in_w, dec_ca_in_b, dec_ca_out_w, dec_ca_out_b,
      dec_ln2_g, dec_ln2_b,
      dec_ff1_w, dec_ff1_b, dec_ff2_w, dec_ff2_b,
      dec_ln3_g, dec_ln3_b,
      out_w, out_b, (float*)d_out);
}